// OTFL_61950608277659
// MI455X (gfx1250) — compile-verified
//
#include <hip/hip_runtime.h>
#include <math.h>

typedef __attribute__((ext_vector_type(2))) float v2f;
typedef __attribute__((ext_vector_type(8))) float v8f;

// Workspace layout (bytes)
#define WSW_OFF   0          // float2[7*256*32]  = 458752 B  (swizzled, zero-padded W fragments)
#define M_OFF     458752     // float[112*112]    = 50176 B   (Gram matrix W^T W, padded)
#define G_OFF     508928     // float[100*112]    = 44800 B   (per-class sum of g in class space)
#define T_OFF     553728     // float[112]        = 448 B     (total sum of g)
#define KEYS_OFF  554176     // u64[100]                      (argmin-CE keys)
#define CNT_OFF   554976     // int[100]
#define CES_OFF   555376     // float                          (sum of CE)
#define TRIP_OFF  555380     // float                          (sum of triplet terms)

__global__ __launch_bounds__(256) void init_kernel(float* __restrict__ G, int* __restrict__ counts,
                                                   unsigned long long* __restrict__ keys,
                                                   float* __restrict__ ces, float* __restrict__ trip) {
  int idx = blockIdx.x * blockDim.x + threadIdx.x;
  int stride = gridDim.x * blockDim.x;
  for (int i = idx; i < 100 * 112; i += stride) G[i] = 0.f;
  for (int i = idx; i < 100; i += stride) { counts[i] = 0; keys[i] = ~0ULL; }
  if (idx == 0) { *ces = 0.f; *trip = 0.f; }
}

// Swizzle W [1024 x 100] into per-tile WMMA B/A fragment order, zero-padding cols 100..111.
// Wsw[(n*256 + ks)*32 + lane] = ( W[4ks+2h][16n+l], W[4ks+2h+1][16n+l] ),  h=lane>>4, l=lane&15
__global__ __launch_bounds__(256) void wprep_kernel(const float* __restrict__ W, float2* __restrict__ Wsw) {
  int tid = blockIdx.x * 256 + threadIdx.x;   // [0, 57344)
  int lane = tid & 31;
  int ks = (tid >> 5) & 255;
  int n = tid >> 13;
  int h = lane >> 4, l = lane & 15;
  int col = n * 16 + l;
  int d0 = 4 * ks + 2 * h;
  float2 v;
  v.x = (col < 100) ? W[d0 * 100 + col] : 0.f;
  v.y = (col < 100) ? W[(d0 + 1) * 100 + col] : 0.f;
  Wsw[tid] = v;
}

// Gram matrix M = Wp^T Wp  (112x112, K=1024) with f32 WMMA. One wave per 16x16 tile.
__global__ __launch_bounds__(32) void gram_kernel(const float2* __restrict__ Wsw, float* __restrict__ M) {
  int ti = blockIdx.x / 7, tj = blockIdx.x % 7;
  int lane = threadIdx.x;
  int h = lane >> 4, l = lane & 15;
  const float2* A = Wsw + ti * 8192 + lane;
  const float2* B = Wsw + tj * 8192 + lane;
  v8f acc = {};
  for (int ks = 0; ks < 256; ++ks) {
    float2 af = A[ks * 32];
    float2 bf = B[ks * 32];
    v2f a; a.x = af.x; a.y = af.y;
    v2f b; b.x = bf.x; b.y = bf.y;
    acc = __builtin_amdgcn_wmma_f32_16x16x4_f32(false, a, false, b, (short)0, acc, false, false);
  }
#pragma unroll
  for (int j = 0; j < 8; ++j)
    M[(ti * 16 + j + 8 * h) * 112 + tj * 16 + l] = acc[j];
}

// Fused: logits = x@W (WMMA f32), softmax, CE, g = p - onehot, segment-sum into G,
// per-class argmin(CE) keys, CE sum, class counts. One block = 16 rows, 4 waves x 2 col-tiles.
__global__ __launch_bounds__(128) void main_kernel(const float* __restrict__ x,
                                                   const float2* __restrict__ Wsw,
                                                   const int* __restrict__ y,
                                                   float* __restrict__ G,
                                                   unsigned long long* __restrict__ keys,
                                                   int* __restrict__ counts,
                                                   float* __restrict__ ce_sum) {
  __shared__ float ls[16][116];
  int r0 = blockIdx.x * 16;
  int tid = threadIdx.x;
  int wave = tid >> 5, lane = tid & 31;
  int h = lane >> 4, l = lane & 15;
  int n0 = wave * 2, n1 = n0 + 1;
  const float2* xa = reinterpret_cast<const float2*>(x + (size_t)(r0 + l) * 1024) + h;
  const float2* wb0 = Wsw + n0 * 8192 + lane;
  const float2* wb1 = Wsw + n1 * 8192 + lane;   // unused when n1 == 7
  v8f acc0 = {}; v8f acc1 = {};
  if (n1 < 7) {                                  // wave-uniform branch: EXEC stays all-ones for WMMA
    for (int ks = 0; ks < 256; ++ks) {
      float2 af = xa[2 * ks];
      v2f a; a.x = af.x; a.y = af.y;
      float2 b0f = wb0[ks * 32];
      float2 b1f = wb1[ks * 32];
      v2f b0; b0.x = b0f.x; b0.y = b0f.y;
      v2f b1; b1.x = b1f.x; b1.y = b1f.y;
      acc0 = __builtin_amdgcn_wmma_f32_16x16x4_f32(false, a, false, b0, (short)0, acc0, false, false);
      acc1 = __builtin_amdgcn_wmma_f32_16x16x4_f32(false, a, false, b1, (short)0, acc1, false, false);
    }
  } else {
    for (int ks = 0; ks < 256; ++ks) {
      float2 af = xa[2 * ks];
      v2f a; a.x = af.x; a.y = af.y;
      float2 b0f = wb0[ks * 32];
      v2f b0; b0.x = b0f.x; b0.y = b0f.y;
      acc0 = __builtin_amdgcn_wmma_f32_16x16x4_f32(false, a, false, b0, (short)0, acc0, false, false);
    }
  }
#pragma unroll
  for (int j = 0; j < 8; ++j) {
    int Mr = j + 8 * h;
    ls[Mr][n0 * 16 + l] = acc0[j];
    if (n1 < 7) ls[Mr][n1 * 16 + l] = acc1[j];
  }
  __syncthreads();

  // Softmax / CE / g accumulation: 8 threads per row.
  int r = tid >> 3, sub = tid & 7;
  float mx = -1e30f;
  for (int c = sub; c < 100; c += 8) mx = fmaxf(mx, ls[r][c]);
#pragma unroll
  for (int o = 1; o < 8; o <<= 1) mx = fmaxf(mx, __shfl_xor(mx, o, 8));
  float se = 0.f;
  for (int c = sub; c < 100; c += 8) se += expf(ls[r][c] - mx);
#pragma unroll
  for (int o = 1; o < 8; o <<= 1) se += __shfl_xor(se, o, 8);
  int cls = y[r0 + r];
  float inv = 1.f / se;
  for (int c = sub; c < 100; c += 8) {
    float g = expf(ls[r][c] - mx) * inv - ((c == cls) ? 1.f : 0.f);
    atomicAdd(&G[cls * 112 + c], g);
  }
  if (sub == 0) {
    float ce = mx + logf(se) - ls[r][cls];
    atomicAdd(ce_sum, ce);
    atomicAdd(&counts[cls], 1);
    unsigned long long key = ((unsigned long long)__float_as_uint(ce) << 32) | (unsigned int)(r0 + r);
    atomicMin(&keys[cls], key);   // ce >= 0 so float-bit order == numeric order; ties -> lowest row
  }
}

__global__ __launch_bounds__(128) void total_kernel(const float* __restrict__ G, float* __restrict__ T) {
  int t = threadIdx.x;
  if (t < 112) {
    float s = 0.f;
    for (int k = 0; k < 100; ++k) s += G[k * 112 + t];
    T[t] = s;
  }
}

// One block per class: recompute anchor g, evaluate quadratic forms against Gram M, add term.
__global__ __launch_bounds__(128) void class_kernel(const float* __restrict__ x,
                                                    const float* __restrict__ W,
                                                    const float* __restrict__ M,
                                                    const float* __restrict__ G,
                                                    const float* __restrict__ T,
                                                    const unsigned long long* __restrict__ keys,
                                                    const int* __restrict__ counts,
                                                    float* __restrict__ trip) {
  __shared__ float sv[128];
  __shared__ float sa[112], sp[112], sn[112];
  int c = blockIdx.x, t = threadIdx.x;
  if (counts[c] == 0) return;                       // uniform across block
  int b = (int)(keys[c] & 0xffffffffULL);
  float lt = -1e30f;
  if (t < 100) {
    const float* xr = x + (size_t)b * 1024;
    float s = 0.f;
    for (int d = 0; d < 1024; ++d) s += xr[d] * W[d * 100 + t];
    lt = s;
  }
  sv[t] = lt; __syncthreads();
  for (int o = 64; o > 0; o >>= 1) { if (t < o) sv[t] = fmaxf(sv[t], sv[t + o]); __syncthreads(); }
  float mx = sv[0]; __syncthreads();
  float e = (t < 100) ? expf(lt - mx) : 0.f;
  sv[t] = e; __syncthreads();
  for (int o = 64; o > 0; o >>= 1) { if (t < o) sv[t] += sv[t + o]; __syncthreads(); }
  float se = sv[0]; __syncthreads();
  if (t < 112) {
    float av = (t < 100) ? (e / se - ((t == c) ? 1.f : 0.f)) : 0.f;
    float gv = G[c * 112 + t];
    sa[t] = av;            // anchor g
    sp[t] = gv - av;       // positives sum minus anchor
    sn[t] = T[t] - gv;     // negatives sum
  }
  __syncthreads();
  float Ma = 0.f, Mp = 0.f, Mn = 0.f;
  if (t < 112) {
    const float* Mr = M + t * 112;
    for (int j = 0; j < 112; ++j) {
      float m = Mr[j];
      Ma += m * sa[j]; Mp += m * sp[j]; Mn += m * sn[j];
    }
  }
  float va = (t < 112) ? sa[t] : 0.f;
  float vp = (t < 112) ? sp[t] : 0.f;
  float vn = (t < 112) ? sn[t] : 0.f;
  float qs[5] = { va * Ma, va * Mp, va * Mn, vp * Mp, vn * Mn };
  float rr[5];
  for (int i = 0; i < 5; ++i) {
    sv[t] = qs[i]; __syncthreads();
    for (int o = 64; o > 0; o >>= 1) { if (t < o) sv[t] += sv[t + o]; __syncthreads(); }
    rr[i] = sv[0]; __syncthreads();
  }
  if (t == 0) {
    const float EPSN = 1e-12f;
    float na = fmaxf(sqrtf(fmaxf(rr[0], 0.f)), EPSN);
    float np = fmaxf(sqrtf(fmaxf(rr[3], 0.f)), EPSN);
    float nn = fmaxf(sqrtf(fmaxf(rr[4], 0.f)), EPSN);
    float term = rr[1] / (na * np) - rr[2] / (na * nn);
    atomicAdd(trip, term);
  }
}

__global__ __launch_bounds__(128) void final_kernel(const int* __restrict__ counts,
                                                    const float* __restrict__ ce_sum,
                                                    const float* __restrict__ trip,
                                                    float* __restrict__ out) {
  __shared__ int si[128];
  int t = threadIdx.x;
  si[t] = (t < 100 && counts[t] > 0) ? 1 : 0;
  __syncthreads();
  for (int o = 64; o > 0; o >>= 1) { if (t < o) si[t] += si[t + o]; __syncthreads(); }
  if (t == 0) {
    float npres = (float)((si[0] > 1) ? si[0] : 1);
    out[0] = ce_sum[0] / 16384.f - 2.0f * (trip[0] / npres);
  }
}

extern "C" void kernel_launch(void* const* d_in, const int* in_sizes, int n_in,
                              void* d_out, int out_size, void* d_ws, size_t ws_size,
                              hipStream_t stream) {
  const float* x = (const float*)d_in[0];   // [16384,1024] f32
  const float* W = (const float*)d_in[1];   // [1024,100]  f32
  const int*   y = (const int*)d_in[2];     // [16384]     i32
  char* ws = (char*)d_ws;
  float2* Wsw = (float2*)(ws + WSW_OFF);
  float* M    = (float*)(ws + M_OFF);
  float* G    = (float*)(ws + G_OFF);
  float* T    = (float*)(ws + T_OFF);
  unsigned long long* keys = (unsigned long long*)(ws + KEYS_OFF);
  int*   counts = (int*)(ws + CNT_OFF);
  float* ces  = (float*)(ws + CES_OFF);
  float* trip = (float*)(ws + TRIP_OFF);

  init_kernel<<<44, 256, 0, stream>>>(G, counts, keys, ces, trip);
  wprep_kernel<<<224, 256, 0, stream>>>(W, Wsw);
  gram_kernel<<<49, 32, 0, stream>>>(Wsw, M);
  main_kernel<<<1024, 128, 0, stream>>>(x, Wsw, y, G, keys, counts, ces);
  total_kernel<<<1, 128, 0, stream>>>(G, T);
  class_kernel<<<100, 128, 0, stream>>>(x, W, M, G, T, keys, counts, trip);
  final_kernel<<<1, 128, 0, stream>>>(counts, ces, trip, (float*)d_out);
}